// gradsym_minibatch_19026705121509
// MI455X (gfx1250) — compile-verified
//
#include <hip/hip_runtime.h>

typedef __attribute__((ext_vector_type(2))) float v2f;
typedef __attribute__((ext_vector_type(8))) float v8f;

#define LSTRIDE 132   // floats per LDS row: 128 data + pad (16B-aligned rows, bank skew)
#define NROWS   108   // 3 components * 6 z-halo * 6 y-halo rows

__device__ __forceinline__ float wave_merge_halves(float lo, float hi, int lane) {
  // lanes 16..31 pick up hi[lane-16]; lanes 0..15 keep lo[lane]
  float h = __int_as_float(
      __builtin_amdgcn_ds_bpermute((lane & 15) << 2, __float_as_int(hi)));
  return (lane < 16) ? lo : h;
}

__global__ __launch_bounds__(256)
void gradsym_minibatch_kernel(const float* __restrict__ u,
                              const float* __restrict__ kern,
                              float* __restrict__ out) {
  __shared__ __align__(16) float lds[NROWS * LSTRIDE];   // 57,024 B

  const int lane = (int)(threadIdx.x & 31u);
  const int wave = (int)(threadIdx.x >> 5u);
  const int m16  = lane & 15;              // WMMA row (A) / column (B) slot
  const int kh   = (lane >> 4) << 1;       // K half: 0 (lanes 0-15) or 2 (lanes 16-31)

  const int y0 = (int)(blockIdx.x << 2);
  const int z0 = (int)(blockIdx.y << 2);
  const int b  = (int)blockIdx.z;

  // ---------- Phase 1: async staging, global -> LDS ----------
  // 108 rows of 128 floats (periodic wrap in y,z via index arithmetic).
  // One row == 32 lanes * 16 B == one global_load_async_to_lds_b128 per wave.
  for (int r = wave; r < NROWS; r += 8) {
    int i  = r / 36;
    int rr = r - i * 36;
    int dz = rr / 6;
    int dy = rr - dz * 6;
    unsigned zz   = (unsigned)(z0 + dz + 127) & 127u;
    unsigned yy   = (unsigned)(y0 + dy + 127) & 127u;
    unsigned chan = (unsigned)(b * 3 + i);
    unsigned goff = ((chan << 21) + (zz << 14) + (yy << 7) + ((unsigned)lane << 2)) << 2;
    unsigned laddr = (unsigned)(unsigned long long)(&lds[r * LSTRIDE + (lane << 2)]);
    asm volatile("global_load_async_to_lds_b128 %0, %1, %2"
                 :: "v"(laddr), "v"(goff), "s"(u)
                 : "memory");
  }
  asm volatile("s_wait_asynccnt 0" ::: "memory");
  __syncthreads();

  // ---------- Weight matrix A (16x28): rows 0..2 = the 3 derivative kernels ----------
  // f32 16x4 A layout: lanes 0-15 hold K=4s,4s+1; lanes 16-31 hold K=4s+2,4s+3.
  v2f A[7];
  #pragma unroll
  for (int s = 0; s < 7; ++s) {
    int t0 = 4 * s + kh;
    float ax = 0.0f, ay = 0.0f;
    if (m16 < 3) {
      ax = (t0     < 27) ? kern[m16 * 27 + t0]     : 0.0f;
      ay = (t0 + 1 < 27) ? kern[m16 * 27 + t0 + 1] : 0.0f;
    }
    A[s].x = ax;
    A[s].y = ay;
  }

  // ---------- Phase 2: WMMA stencil over the 4x4x128 tile ----------
  // 64 wave-iterations of 32 x-points each, distributed over 8 waves.
  for (int it = wave; it < 64; it += 8) {
    int lz  = it >> 4;
    int ly  = (it >> 2) & 3;
    int xg0 = (it & 3) << 5;

    float d[2][3][3];

    #pragma unroll
    for (int grp = 0; grp < 2; ++grp) {
      int xb = xg0 + (grp << 4) + m16;     // this lane's B column (x-point)
      #pragma unroll
      for (int i = 0; i < 3; ++i) {
        int rowbase = i * 36 + lz * 6 + ly;
        v8f c = {};
        #pragma unroll
        for (int s = 0; s < 7; ++s) {
          int t0 = 4 * s + kh;
          // tap t = dz*9 + dy*3 + dx, offsets -1..1 with wrap in x via &127
          int ta = t0;
          int za = ta / 9; int ra = ta - za * 9; int ya = ra / 3; int xa = ra - ya * 3;
          float bx = lds[(rowbase + za * 6 + ya) * LSTRIDE + ((xb + xa + 127) & 127)];
          // tail tap: branch-free (clamped unconditional load + VALU select)
          int tb  = t0 + 1;
          int tbc = (tb < 27) ? tb : 26;
          int zb = tbc / 9; int rb = tbc - zb * 9; int yb = rb / 3; int xc = rb - yb * 3;
          float byr = lds[(rowbase + zb * 6 + yb) * LSTRIDE + ((xb + xc + 127) & 127)];
          float by  = (tb < 27) ? byr : 0.0f;
          v2f bv; bv.x = bx; bv.y = by;
          // D(16x16) += A(16x4) x B(4x16): rows = kernels, cols = x-points
          c = __builtin_amdgcn_wmma_f32_16x16x4_f32(false, A[s], false, bv,
                                                    (short)0, c, false, false);
        }
        // C/D layout: VGPR r holds row M=r in lanes 0-15 -> d_j for 16 points
        d[grp][i][0] = c[0];
        d[grp][i][1] = c[1];
        d[grp][i][2] = c[2];
      }
    }

    // Merge the two 16-point groups so lane L holds x = xg0 + L (full 128B stores)
    float dm[3][3];
    #pragma unroll
    for (int i = 0; i < 3; ++i)
      #pragma unroll
      for (int j = 0; j < 3; ++j)
        dm[i][j] = wave_merge_halves(d[0][i][j], d[1][i][j], lane);

    float xx = dm[0][0];
    float yy = dm[1][1];
    float zz = dm[2][2];
    float xy = 0.5f * (dm[0][1] + dm[1][0]);
    float xz = 0.5f * (dm[0][2] + dm[2][0]);
    float yz = 0.5f * (dm[1][2] + dm[2][1]);

    size_t sp = ((size_t)(z0 + lz) << 14) + ((size_t)(y0 + ly) << 7)
              + (size_t)(xg0 + lane);

    // out[0, g*4+b, m, z, y, x]  ->  field index f = (g*4+b)*3 + m
    // Non-temporal: 288 MiB of streaming stores must not evict the 96 MiB
    // input working set from the 192 MB L2 (halo reuse lives there).
    auto st = [&](int g, int m, float v) {
      __builtin_nontemporal_store(
          v, out + (((size_t)((g * 4 + b) * 3 + m)) << 21) + sp);
    };
    st(0, 0, xx); st(0, 1, xy); st(0, 2, xz);
    st(1, 0, xy); st(1, 1, yy); st(1, 2, yz);
    st(2, 0, xz); st(2, 1, yz); st(2, 2, zz);
  }
}

extern "C" void kernel_launch(void* const* d_in, const int* in_sizes, int n_in,
                              void* d_out, int out_size, void* d_ws, size_t ws_size,
                              hipStream_t stream) {
  (void)in_sizes; (void)n_in; (void)out_size; (void)d_ws; (void)ws_size;
  const float* u    = (const float*)d_in[0];
  const float* kern = (const float*)d_in[1];
  float* out        = (float*)d_out;
  dim3 grid(32, 32, 4);   // y-tiles, z-tiles, batch
  dim3 block(256, 1, 1);  // 8 waves
  hipLaunchKernelGGL(gradsym_minibatch_kernel, grid, block, 0, stream, u, kern, out);
}